// NGCF_70196945486132
// MI455X (gfx1250) — compile-verified
//
#include <hip/hip_runtime.h>

// ---------------- problem constants (match reference) ----------------
#define N_USERS 100000
#define N_ITEMS 50000
#define N_NODES (N_USERS + N_ITEMS)   // 150000 (divisible by 16)
#define D       64
#define N_LAYERS 3
#define OUT_COLS (D * (N_LAYERS + 1)) // 256

typedef __attribute__((ext_vector_type(2))) float v2f;
typedef __attribute__((ext_vector_type(8))) float v8f;

// ---------------------------------------------------------------------
// Kernel 1: copy initial embeddings into ego workspace AND out[:, 0:64]
// (layer-0 slice of the output is the UN-normalized initial embedding)
// ---------------------------------------------------------------------
__global__ __launch_bounds__(256) void ngcf_init(
    const float* __restrict__ user_emb, const float* __restrict__ item_emb,
    float* __restrict__ ego, float* __restrict__ out)
{
    int idx = blockIdx.x * blockDim.x + threadIdx.x;   // one float4 group
    if (idx >= N_NODES * (D / 4)) return;
    int node = idx >> 4;           // D/4 == 16 groups per node
    int g    = idx & 15;
    const float4* src = (node < N_USERS)
        ? (const float4*)(user_emb) + node * 16 + g
        : (const float4*)(item_emb) + (node - N_USERS) * 16 + g;
    float4 v = *src;
    ((float4*)(ego + node * D))[g] = v;
    ((float4*)(out + (size_t)node * OUT_COLS))[g] = v;  // columns [0:64)
}

// ---------------------------------------------------------------------
// Kernel 2: COO SpMM  side[r] += vals[e] * ego[cols[e]]
// 16 threads per edge, float4 gather (L2-resident), 4 atomics each.
// ---------------------------------------------------------------------
__global__ __launch_bounds__(256) void ngcf_spmm(
    const float* __restrict__ vals, const int* __restrict__ rows,
    const int* __restrict__ cols, const float* __restrict__ x,
    float* __restrict__ y, int nnz)
{
    int t = blockIdx.x * blockDim.x + threadIdx.x;
    int e = t >> 4;
    if (e >= nnz) return;
    int d4 = (t & 15) * 4;
    float v = vals[e];
    int c = cols[e];
    int r = rows[e];
    float4 xv = *(const float4*)(x + (size_t)c * D + d4);
    float* yp = y + (size_t)r * D + d4;
    atomicAdd(yp + 0, v * xv.x);
    atomicAdd(yp + 1, v * xv.y);
    atomicAdd(yp + 2, v * xv.z);
    atomicAdd(yp + 3, v * xv.w);
}

// ---------------------------------------------------------------------
// Kernel 3: dense layer, one wave32 per 16-row tile.
//   sum_e = lrelu(side @ gc_w^T + gc_b)
//   bi    = lrelu((ego*side) @ bi_w^T + bi_b)
//   ego   = sum_e + bi   (in place; wave reads its rows before writing)
// Uses V_WMMA_F32_16X16X4_F32: 4 col-tiles x 16 k-steps x 2 GEMMs.
// A-fragment (16x4 f32): lane(hi,lo) holds A[lo][4s+2hi], A[lo][4s+2hi+1]
// B-fragment (4x16 f32): lane(hi,lo) holds B[4s+2hi][lo], B[4s+2hi+1][lo]
//   with B = W^T  =>  float2 at W[(jt*16+lo)*64 + 4s+2hi]
// ---------------------------------------------------------------------
__global__ __launch_bounds__(256) void ngcf_dense_wmma(
    const float* __restrict__ side, float* __restrict__ ego,
    const float* __restrict__ gcw, const float* __restrict__ gcb,
    const float* __restrict__ biw, const float* __restrict__ bib)
{
    const int wave = (blockIdx.x * blockDim.x + threadIdx.x) >> 5;
    const int lane = threadIdx.x & 31;
    if (wave >= N_NODES / 16) return;
    const int row0 = wave * 16;
    const int hi = lane >> 4;      // 0 or 1
    const int lo = lane & 15;

    // Preload this wave's A-fragments for all 16 k-steps (each element once).
    v2f sideF[16], egoF[16];
    const float* srow = side + (size_t)(row0 + lo) * D + 2 * hi;
    const float* erow = ego  + (size_t)(row0 + lo) * D + 2 * hi;
#pragma unroll
    for (int s = 0; s < 16; ++s) {
        sideF[s] = *(const v2f*)(srow + 4 * s);
        egoF[s]  = *(const v2f*)(erow + 4 * s);
    }

    float egoNew[4][8];
#pragma unroll
    for (int jt = 0; jt < 4; ++jt) {
        const int j = jt * 16 + lo;                  // output column this lane owns
        const float* gwp = gcw + (size_t)j * D + 2 * hi;
        const float* bwp = biw + (size_t)j * D + 2 * hi;

        v8f accG = {};
#pragma unroll
        for (int s = 0; s < 16; ++s) {
            v2f bg = *(const v2f*)(gwp + 4 * s);
            accG = __builtin_amdgcn_wmma_f32_16x16x4_f32(
                false, sideF[s], false, bg, (short)0, accG, false, false);
        }
        v8f accB = {};
#pragma unroll
        for (int s = 0; s < 16; ++s) {
            v2f prod = egoF[s] * sideF[s];           // (ego*side) A-fragment
            v2f bb = *(const v2f*)(bwp + 4 * s);
            accB = __builtin_amdgcn_wmma_f32_16x16x4_f32(
                false, prod, false, bb, (short)0, accB, false, false);
        }

        const float bgc = gcb[j];
        const float bbi = bib[j];
#pragma unroll
        for (int v = 0; v < 8; ++v) {
            float x1 = accG[v] + bgc; x1 = (x1 > 0.0f) ? x1 : 0.01f * x1;
            float x2 = accB[v] + bbi; x2 = (x2 > 0.0f) ? x2 : 0.01f * x2;
            egoNew[jt][v] = x1 + x2;
        }
    }

    // D layout: vgpr v, lanes 0-15 -> row v; lanes 16-31 -> row v+8; col = lo
#pragma unroll
    for (int jt = 0; jt < 4; ++jt)
#pragma unroll
        for (int v = 0; v < 8; ++v)
            ego[(size_t)(row0 + v + 8 * hi) * D + jt * 16 + lo] = egoNew[jt][v];
}

// ---------------------------------------------------------------------
// Kernel 4: row L2-normalize ego into out[:, colbase : colbase+64)
// One wave per row; lane holds float2; shuffle reduction (wave32).
// ---------------------------------------------------------------------
__global__ __launch_bounds__(256) void ngcf_normalize(
    const float* __restrict__ ego, float* __restrict__ out, int colbase)
{
    int wave = (blockIdx.x * blockDim.x + threadIdx.x) >> 5;
    int lane = threadIdx.x & 31;
    if (wave >= N_NODES) return;
    float2 v = *(const float2*)(ego + (size_t)wave * D + lane * 2);
    float ss = v.x * v.x + v.y * v.y;
#pragma unroll
    for (int off = 16; off >= 1; off >>= 1)
        ss += __shfl_xor(ss, off, 32);
    float inv = 1.0f / fmaxf(sqrtf(ss), 1e-12f);
    float2 o; o.x = v.x * inv; o.y = v.y * inv;
    *(float2*)(out + (size_t)wave * OUT_COLS + colbase + lane * 2) = o;
}

// ---------------------------------------------------------------------
extern "C" void kernel_launch(void* const* d_in, const int* in_sizes, int n_in,
                              void* d_out, int out_size, void* d_ws, size_t ws_size,
                              hipStream_t stream) {
    const float* user_emb = (const float*)d_in[0];
    const float* item_emb = (const float*)d_in[1];
    const float* gc_w     = (const float*)d_in[2];   // [3,64,64]
    const float* gc_b     = (const float*)d_in[3];   // [3,64]
    const float* bi_w     = (const float*)d_in[4];
    const float* bi_b     = (const float*)d_in[5];
    const float* adj_vals = (const float*)d_in[6];
    const int*   adj_rows = (const int*)d_in[7];
    const int*   adj_cols = (const int*)d_in[8];
    const int nnz = in_sizes[6];

    float* out  = (float*)d_out;
    float* ego  = (float*)d_ws;                          // [N_NODES, 64]
    float* side = ego + (size_t)N_NODES * D;             // [N_NODES, 64]

    // init: ego workspace + layer-0 output slice
    {
        int total = N_NODES * (D / 4);
        ngcf_init<<<(total + 255) / 256, 256, 0, stream>>>(user_emb, item_emb, ego, out);
    }

    for (int layer = 0; layer < N_LAYERS; ++layer) {
        hipMemsetAsync(side, 0, (size_t)N_NODES * D * sizeof(float), stream);

        {
            long long threads = (long long)nnz * 16;
            int blocks = (int)((threads + 255) / 256);
            ngcf_spmm<<<blocks, 256, 0, stream>>>(adj_vals, adj_rows, adj_cols,
                                                  ego, side, nnz);
        }
        {
            int waves = N_NODES / 16;                      // 9375 tiles
            int blocks = (waves * 32 + 255) / 256;
            ngcf_dense_wmma<<<blocks, 256, 0, stream>>>(
                side, ego,
                gc_w + (size_t)layer * D * D, gc_b + (size_t)layer * D,
                bi_w + (size_t)layer * D * D, bi_b + (size_t)layer * D);
        }
        {
            int blocks = (N_NODES * 32 + 255) / 256;
            ngcf_normalize<<<blocks, 256, 0, stream>>>(ego, out, (layer + 1) * D);
        }
    }
}